// MoELayer_10350871183503
// MI455X (gfx1250) — compile-verified
//
#include <hip/hip_runtime.h>
#include <hip/hip_bf16.h>

// ---------------- static MoE config (matches reference) ----------------
#define NE   8          // experts
#define NK   2          // top-k
#define NB   4
#define NS   2048
#define ND   1024
#define NFF  4096
#define NT   (NB * NS)      // 8192 tokens
#define NCAP 1280           // floor(NT/NE * 1.25)
#define AUXC 0.01f

typedef __attribute__((ext_vector_type(2))) float v2f;
typedef __attribute__((ext_vector_type(8))) float v8f;

#define LDST 36   // LDS row stride (floats): 36*r mod 64 distinct for r=0..15 -> conflict free, 8B aligned

// ---------------- workspace layout ----------------
static constexpr size_t OFF_IDX  = 0;                                      // [NT*NK] int
static constexpr size_t OFF_POS  = OFF_IDX + (size_t)NT * NK * 4;          // [NT*NK] int
static constexpr size_t OFF_RW   = OFF_POS + (size_t)NT * NK * 4;          // [NT*NK] f32
static constexpr size_t OFF_AUX  = OFF_RW  + (size_t)NT * NK * 4;          // [16] f32: probs_sum[0..7], counts_sum[8..15]
static constexpr size_t OFF_BUF  = (OFF_AUX + 16 * 4 + 255) & ~(size_t)255;        // [NE,NCAP,ND] f32
static constexpr size_t OFF_H    = OFF_BUF + (size_t)NE * NCAP * ND * 4;           // [NE,NCAP,NFF] f32
static constexpr size_t OFF_EOUT = OFF_H   + (size_t)NE * NCAP * NFF * 4;          // [NE,NCAP,ND] f32

// ---------------- helpers ----------------
__global__ __launch_bounds__(256) void zero_kernel(float* __restrict__ p, size_t n) {
    size_t i = (size_t)blockIdx.x * blockDim.x + threadIdx.x;
    size_t stride = (size_t)gridDim.x * blockDim.x;
    for (; i < n; i += stride) p[i] = 0.0f;
}

// ---------------- 1) router: logits, top-2, softmaxes, aux accumulators ----------------
__global__ __launch_bounds__(256) void router_kernel(
    const float* __restrict__ xf, const float* __restrict__ gate_w,
    int* __restrict__ idx, float* __restrict__ rw, float* __restrict__ auxbuf)
{
    __shared__ float sg[NE * ND];        // 32 KB gate weights
    __shared__ float s_probs[NE];
    __shared__ float s_counts[NE];
    const int tid = threadIdx.x;
    for (int i = tid; i < NE * ND; i += 256) sg[i] = gate_w[i];
    if (tid < NE) { s_probs[tid] = 0.0f; s_counts[tid] = 0.0f; }
    __syncthreads();

    const int wave = tid >> 5, lane = tid & 31;
    const int t = blockIdx.x * 8 + wave;       // one wave32 per token
    float acc[NE];
#pragma unroll
    for (int e = 0; e < NE; ++e) acc[e] = 0.0f;
    const float* xr = xf + (size_t)t * ND;
    for (int d = lane; d < ND; d += 32) {
        float xv = xr[d];
#pragma unroll
        for (int e = 0; e < NE; ++e) acc[e] = fmaf(xv, sg[e * ND + d], acc[e]);
    }
#pragma unroll
    for (int e = 0; e < NE; ++e) {
        float v = acc[e];
        for (int off = 16; off > 0; off >>= 1) v += __shfl_xor(v, off, 32);
        acc[e] = v;
    }
    if (lane == 0) {
        // top-2 (ties -> lowest index, matching jax top_k)
        int i0 = 0; float v0 = acc[0];
        for (int e = 1; e < NE; ++e) if (acc[e] > v0) { v0 = acc[e]; i0 = e; }
        int i1 = (i0 == 0) ? 1 : 0; float v1 = acc[i1];
        for (int e = 0; e < NE; ++e) if (e != i0 && acc[e] > v1) { v1 = acc[e]; i1 = e; }
        // softmax over the two raw top values
        float e0 = expf(v0 - v0), e1 = expf(v1 - v0);
        float inv = 1.0f / (e0 + e1);
        idx[t * 2] = i0; idx[t * 2 + 1] = i1;
        rw[t * 2] = e0 * inv; rw[t * 2 + 1] = e1 * inv;
        // full softmax for aux loss
        float mx = acc[0];
        for (int e = 1; e < NE; ++e) mx = fmaxf(mx, acc[e]);
        float pe[NE]; float se = 0.0f;
        for (int e = 0; e < NE; ++e) { pe[e] = expf(acc[e] - mx); se += pe[e]; }
        float ise = 1.0f / se;
        for (int e = 0; e < NE; ++e) atomicAdd(&s_probs[e], pe[e] * ise);
        atomicAdd(&s_counts[i0], 1.0f);
        atomicAdd(&s_counts[i1], 1.0f);
    }
    __syncthreads();
    if (tid < NE) {
        atomicAdd(&auxbuf[tid],     s_probs[tid]);
        atomicAdd(&auxbuf[8 + tid], s_counts[tid]);
    }
}

// ---------------- 2) ordered per-expert positions (token-major cumsum order) ----------------
__global__ __launch_bounds__(256) void pos_kernel(const int* __restrict__ idx, int* __restrict__ pos)
{
    __shared__ int cnt[256][NE];   // 8 KB
    const int t = threadIdx.x;
    const int base = t * ((NT * NK) / 256);     // 64 consecutive flat items each
    int c[NE];
#pragma unroll
    for (int e = 0; e < NE; ++e) c[e] = 0;
    for (int i = 0; i < (NT * NK) / 256; ++i) c[idx[base + i]]++;
#pragma unroll
    for (int e = 0; e < NE; ++e) cnt[t][e] = c[e];
    __syncthreads();
    int off[NE];
#pragma unroll
    for (int e = 0; e < NE; ++e) off[e] = 0;
    for (int p = 0; p < t; ++p)
#pragma unroll
        for (int e = 0; e < NE; ++e) off[e] += cnt[p][e];
    for (int i = 0; i < (NT * NK) / 256; ++i) {
        int e = idx[base + i];
        pos[base + i] = off[e]++;
    }
}

// ---------------- 3) dispatch scatter into [NE, NCAP, ND] ----------------
__global__ __launch_bounds__(256) void dispatch_kernel(
    const float* __restrict__ xf, const int* __restrict__ idx,
    const int* __restrict__ pos, float* __restrict__ buf)
{
    const int i = blockIdx.x;           // flat (t,k)
    const int p = pos[i];
    if (p >= NCAP) return;              // capacity-dropped: buf stays zero
    const int e = idx[i];
    const float4* src = (const float4*)(xf + (size_t)(i >> 1) * ND);
    float4* dst = (float4*)(buf + ((size_t)e * NCAP + p) * ND);
    dst[threadIdx.x] = src[threadIdx.x];   // 256 * float4 = 1024 floats
}

// ---------------- WMMA tile helpers ----------------
__device__ __forceinline__ void load_tile64x32(const float* __restrict__ g, int ldg,
                                               float* __restrict__ lds, int tid)
{
    // 64 rows x 32 cols, 256 threads: 4 threads/row, 8 floats each
    const int r = tid >> 2;
    const int c = (tid & 3) * 8;
    const float* src = g + (size_t)r * ldg + c;
    float* d = lds + r * LDST + c;
    float4 v0 = *(const float4*)(src);
    float4 v1 = *(const float4*)(src + 4);
    d[0] = v0.x; d[1] = v0.y; d[2] = v0.z; d[3] = v0.w;
    d[4] = v1.x; d[5] = v1.y; d[6] = v1.z; d[7] = v1.w;
}

__device__ __forceinline__ v2f frag_ld(const float* __restrict__ lds, int row, int kk, int lane)
{
    // A 16x4 layout: lanes 0-15 hold K={kk,kk+1}, lanes 16-31 hold K={kk+2,kk+3}; B symmetric on N.
    const int koff = (lane >> 4) << 1;
    const float* p = lds + row * LDST + kk + koff;
    v2f r; r.x = p[0]; r.y = p[1];
    return r;
}

__device__ __forceinline__ v8f wmma_f32(v2f a, v2f b, v8f c)
{
    return __builtin_amdgcn_wmma_f32_16x16x4_f32(false, a, false, b, (short)0, c, false, false);
}

// ---------------- 4) fused h = silu(buf @ w1^T) * (buf @ w3^T) ----------------
__global__ __launch_bounds__(256) void gemm13_kernel(
    const float* __restrict__ buf, const float* __restrict__ w1,
    const float* __restrict__ w3, float* __restrict__ h)
{
    __shared__ float lA[64 * LDST], lB1[64 * LDST], lB3[64 * LDST];   // ~27.6 KB
    const int e  = blockIdx.z;
    const int m0 = blockIdx.y * 64;
    const int n0 = blockIdx.x * 64;
    const int tid = threadIdx.x, lane = tid & 31, wave = tid >> 5;
    const int wm = (wave >> 1) * 16;   // 0,16,32,48
    const int wn = (wave & 1) * 32;    // 0,32
    const int row = lane & 15;

    const float* A  = buf + (size_t)e * NCAP * ND + (size_t)m0 * ND;
    const float* B1 = w1  + (size_t)e * NFF  * ND + (size_t)n0 * ND;
    const float* B3 = w3  + (size_t)e * NFF  * ND + (size_t)n0 * ND;

    v8f a10 = {}, a11 = {}, a30 = {}, a31 = {};
    for (int k0 = 0; k0 < ND; k0 += 32) {
        __syncthreads();
        load_tile64x32(A  + k0, ND, lA,  tid);
        load_tile64x32(B1 + k0, ND, lB1, tid);
        load_tile64x32(B3 + k0, ND, lB3, tid);
        __syncthreads();
#pragma unroll
        for (int kk = 0; kk < 32; kk += 4) {
            v2f a   = frag_ld(lA  + wm * LDST,        row, kk, lane);
            v2f b10 = frag_ld(lB1 + wn * LDST,        row, kk, lane);
            v2f b11 = frag_ld(lB1 + (wn + 16) * LDST, row, kk, lane);
            v2f b30 = frag_ld(lB3 + wn * LDST,        row, kk, lane);
            v2f b31 = frag_ld(lB3 + (wn + 16) * LDST, row, kk, lane);
            a10 = wmma_f32(a, b10, a10);
            a11 = wmma_f32(a, b11, a11);
            a30 = wmma_f32(a, b30, a30);
            a31 = wmma_f32(a, b31, a31);
        }
    }
    // epilogue: SwiGLU in-register, write h
    const int half = lane >> 4;     // C/D: VGPR j holds M=j (lanes 0-15) or M=j+8 (lanes 16-31)
    const int col  = lane & 15;
    float* H = h + (size_t)e * NCAP * NFF + (size_t)(m0 + wm) * NFF + n0 + wn;
#pragma unroll
    for (int j = 0; j < 8; ++j) {
        const int r = j + half * 8;
        float x1 = a10[j], x3 = a30[j];
        H[(size_t)r * NFF + col]      = (x1 / (1.0f + expf(-x1))) * x3;
        float y1 = a11[j], y3 = a31[j];
        H[(size_t)r * NFF + 16 + col] = (y1 / (1.0f + expf(-y1))) * y3;
    }
}

// ---------------- 5) eout = h @ w2^T ----------------
__global__ __launch_bounds__(256) void gemm2_kernel(
    const float* __restrict__ h, const float* __restrict__ w2, float* __restrict__ eout)
{
    __shared__ float lA[64 * LDST], lB[64 * LDST];
    const int e  = blockIdx.z;
    const int m0 = blockIdx.y * 64;
    const int n0 = blockIdx.x * 64;
    const int tid = threadIdx.x, lane = tid & 31, wave = tid >> 5;
    const int wm = (wave >> 1) * 16;
    const int wn = (wave & 1) * 32;
    const int row = lane & 15;

    const float* A  = h  + (size_t)e * NCAP * NFF + (size_t)m0 * NFF;
    const float* Bm = w2 + (size_t)e * ND   * NFF + (size_t)n0 * NFF;

    v8f a0 = {}, a1 = {};
    for (int k0 = 0; k0 < NFF; k0 += 32) {
        __syncthreads();
        load_tile64x32(A  + k0, NFF, lA, tid);
        load_tile64x32(Bm + k0, NFF, lB, tid);
        __syncthreads();
#pragma unroll
        for (int kk = 0; kk < 32; kk += 4) {
            v2f a  = frag_ld(lA + wm * LDST,        row, kk, lane);
            v2f b0 = frag_ld(lB + wn * LDST,        row, kk, lane);
            v2f b1 = frag_ld(lB + (wn + 16) * LDST, row, kk, lane);
            a0 = wmma_f32(a, b0, a0);
            a1 = wmma_f32(a, b1, a1);
        }
    }
    const int half = lane >> 4;
    const int col  = lane & 15;
    float* O = eout + (size_t)e * NCAP * ND + (size_t)(m0 + wm) * ND + n0 + wn;
#pragma unroll
    for (int j = 0; j < 8; ++j) {
        const int r = j + half * 8;
        O[(size_t)r * ND + col]      = a0[j];
        O[(size_t)r * ND + 16 + col] = a1[j];
    }
}

// ---------------- 6) combine: gather + masked/renormalized weighting ----------------
__global__ __launch_bounds__(256) void combine_kernel(
    const float* __restrict__ eout, const int* __restrict__ idx,
    const int* __restrict__ pos, const float* __restrict__ rw,
    float* __restrict__ out)
{
    const int t = blockIdx.x;
    const int i0 = idx[2 * t], i1 = idx[2 * t + 1];
    const int p0 = pos[2 * t], p1 = pos[2 * t + 1];
    const float m0 = (p0 < NCAP) ? 1.0f : 0.0f;
    const float m1 = (p1 < NCAP) ? 1.0f : 0.0f;
    float w0 = rw[2 * t] * m0, w1 = rw[2 * t + 1] * m1;
    const float inv = 1.0f / (w0 + w1 + 1e-8f);
    w0 *= inv; w1 *= inv;
    const int p0c = min(p0, NCAP - 1), p1c = min(p1, NCAP - 1);
    const float4* g0 = (const float4*)(eout + ((size_t)i0 * NCAP + p0c) * ND);
    const float4* g1 = (const float4*)(eout + ((size_t)i1 * NCAP + p1c) * ND);
    float4* o = (float4*)(out + (size_t)t * ND);
    const int c = threadIdx.x;
    float4 a = g0[c], b = g1[c], r;
    r.x = w0 * a.x + w1 * b.x;
    r.y = w0 * a.y + w1 * b.y;
    r.z = w0 * a.z + w1 * b.z;
    r.w = w0 * a.w + w1 * b.w;
    o[c] = r;
}

// ---------------- 7) aux loss scalar ----------------
__global__ void aux_kernel(const float* __restrict__ auxbuf, float* __restrict__ aux_out)
{
    float s = 0.0f;
    for (int e = 0; e < NE; ++e)
        s += (auxbuf[8 + e] / (float)NT) * (auxbuf[e] / (float)NT);   // frac * mean_probs
    *aux_out = AUXC * s * (float)NE;
}

// ---------------- launcher ----------------
extern "C" void kernel_launch(void* const* d_in, const int* in_sizes, int n_in,
                              void* d_out, int out_size, void* d_ws, size_t ws_size,
                              hipStream_t stream)
{
    const float* x      = (const float*)d_in[0];   // [NB,NS,ND]
    const float* gate_w = (const float*)d_in[1];   // [NE,ND]
    const float* w1     = (const float*)d_in[2];   // [NE,NFF,ND]
    const float* w3     = (const float*)d_in[3];   // [NE,NFF,ND]
    const float* w2     = (const float*)d_in[4];   // [NE,ND,NFF]
    float* out = (float*)d_out;                    // [NT*ND] + 1 aux scalar

    char* ws = (char*)d_ws;
    int*   idx  = (int*)(ws + OFF_IDX);
    int*   pos  = (int*)(ws + OFF_POS);
    float* rw   = (float*)(ws + OFF_RW);
    float* aux  = (float*)(ws + OFF_AUX);
    float* buf  = (float*)(ws + OFF_BUF);
    float* h    = (float*)(ws + OFF_H);
    float* eout = (float*)(ws + OFF_EOUT);

    // zero dispatch buffer + aux accumulators (fresh every call)
    zero_kernel<<<2048, 256, 0, stream>>>(buf, (size_t)NE * NCAP * ND);
    zero_kernel<<<1, 256, 0, stream>>>(aux, 16);

    // routing
    router_kernel<<<NT / 8, 256, 0, stream>>>(x, gate_w, idx, rw, aux);
    pos_kernel<<<1, 256, 0, stream>>>(idx, pos);
    dispatch_kernel<<<NT * NK, 256, 0, stream>>>(x, idx, pos, buf);

    // expert FFN (WMMA f32)
    gemm13_kernel<<<dim3(NFF / 64, NCAP / 64, NE), 256, 0, stream>>>(buf, w1, w3, h);
    gemm2_kernel<<<dim3(ND / 64, NCAP / 64, NE), 256, 0, stream>>>(h, w2, eout);

    // combine + aux
    combine_kernel<<<NT, 256, 0, stream>>>(eout, idx, pos, rw, out);
    aux_kernel<<<1, 1, 0, stream>>>(aux, out + (size_t)NT * ND);
}